// FourierKANAdapter_42958262895274
// MI455X (gfx1250) — compile-verified
//
#include <hip/hip_runtime.h>

// FourierKAN correction: out[t,d] = x + c0[d] + sum_k sin(kx)*c_sk[d] + cos(kx)*c_ck[d]
// Memory-bound elementwise stream: 256 MB @ 23.3 TB/s ~= 11 us floor.
// CDNA5 paths exercised: TDM tensor_load_to_lds (TENSORcnt), async global->LDS
// (ASYNCcnt), non-temporal b128 stream, global_prefetch_b8, split barriers.

typedef float v4f __attribute__((ext_vector_type(4)));
typedef unsigned int u32;
typedef u32 v4u __attribute__((ext_vector_type(4)));
typedef u32 v8u __attribute__((ext_vector_type(8)));

#define DCOLS 512
#define TROWS 65536
#define NC 7                       // 2K+1 with K=3
#define TPB 256
#define NBLOCKS 1024
#define CDWORDS (DCOLS * NC)       // 3584 dwords = 14336 bytes
#define CVEC (CDWORDS / 4)         // 896 b128 transfers

__global__ __launch_bounds__(TPB, 2) void fourier_kan_kernel(
    const float* __restrict__ x,
    const float* __restrict__ coeffs,
    float* __restrict__ out)
{
    __shared__ __align__(16) float sc[CDWORDS];   // 14336 bytes

    const int tid = threadIdx.x;
    const unsigned ldsBase = (unsigned)(size_t)(void*)&sc[0];

    // ---- Staging path 1: Tensor Data Mover. Wave 0 issues one descriptor-driven
    // DMA of the whole 14336-byte coefficient tile (1 x 3584 dwords). ----
    if (tid < 32) {
        unsigned long long ga = (unsigned long long)(size_t)coeffs;
        v4u g0;
        g0[0] = 1u;                                   // count=1 (valid), no gather
        g0[1] = ldsBase;                              // lds_addr [63:32]
        g0[2] = (u32)ga;                              // global_addr [95:64]
        g0[3] = ((u32)(ga >> 32) & 0x01FFFFFFu)       // global_addr [120:96]
              | (2u << 30);                           // type=2 ("image") [127:126]
        v8u g1;
        g1[0] = 2u << 16;                             // data_size=2 (4 bytes); mask=0
        g1[1] = (u32)CDWORDS << 16;                   // tensor_dim0[15:0] @ bits 63:48
        g1[2] = 1u << 16;                             // tensor_dim0[31:16]=0 | tensor_dim1=1
        g1[3] = (u32)CDWORDS << 16;                   // tile_dim0=3584 @ bits 127:112
        g1[4] = 1u;                                   // tile_dim1=1, tile_dim2=0
        g1[5] = (u32)CDWORDS;                         // tensor_dim0_stride[31:0]
        g1[6] = 0u;                                   // stride msbs / dim1_stride
        g1[7] = 0u;
        asm volatile("tensor_load_to_lds %0, %1" :: "s"(g0), "s"(g1) : "memory");
        asm volatile("s_wait_tensorcnt 0x0" ::: "memory");
    }

    // ---- Staging path 2: per-lane async global->LDS b128 (same data, same
    // destination -> race-free; both engines read from L2-resident table). ----
    for (int i = tid; i < CVEC; i += TPB) {
        unsigned goff  = (unsigned)i * 16u;           // byte offset into coeffs
        unsigned laddr = ldsBase + goff;              // LDS byte address
        asm volatile("global_load_async_to_lds_b128 %0, %1, %2"
                     :: "v"(laddr), "v"(goff), "s"(coeffs)
                     : "memory");
    }
    asm volatile("s_wait_asynccnt 0x0" ::: "memory");
    __syncthreads();

    // Each thread owns 4 consecutive columns; 128 threads span a row,
    // a 256-thread block covers 2 contiguous rows per step.
    const int cg = tid & 127;     // column group
    const int rh = tid >> 7;      // which row of the pair
    const int d0 = cg * 4;

    // Pull this thread's 28 coefficients (16B-aligned, consecutive) into VGPRs.
    float C[4][NC];
#pragma unroll
    for (int j = 0; j < 4; ++j)
#pragma unroll
        for (int i = 0; i < NC; ++i)
            C[j][i] = sc[(d0 + j) * NC + i];

    const int rowStride = (int)gridDim.x * 2;
    for (int row = (int)blockIdx.x * 2 + rh; row < TROWS; row += rowStride) {
        const int idx = row * DCOLS + d0;

        // Speculative prefetch one grid-stride ahead (global_prefetch_b8).
        __builtin_prefetch(x + idx + rowStride * DCOLS, 0, 0);

        // Streaming (non-temporal) b128 load of 4 elements.
        v4f xv = __builtin_nontemporal_load((const v4f*)(x + idx));
        v4f ov;

#pragma unroll
        for (int j = 0; j < 4; ++j) {
            float xs = xv[j];
            float s1, c1;
            __sincosf(xs, &s1, &c1);                   // v_sin_f32 / v_cos_f32
            float ss = s1 * s1;
            float s2 = 2.0f * s1 * c1;                 // sin 2x
            float c2 = 1.0f - 2.0f * ss;               // cos 2x
            float s3 = s1 * (3.0f - 4.0f * ss);        // sin 3x
            float c3 = c1 * (4.0f * (c1 * c1) - 3.0f); // cos 3x
            float r = C[j][0];
            r = __fmaf_rn(s1, C[j][1], r);
            r = __fmaf_rn(c1, C[j][2], r);
            r = __fmaf_rn(s2, C[j][3], r);
            r = __fmaf_rn(c2, C[j][4], r);
            r = __fmaf_rn(s3, C[j][5], r);
            r = __fmaf_rn(c3, C[j][6], r);
            ov[j] = xs + r;
        }

        // Streaming (non-temporal) b128 store.
        __builtin_nontemporal_store(ov, (v4f*)(out + idx));
    }
}

extern "C" void kernel_launch(void* const* d_in, const int* in_sizes, int n_in,
                              void* d_out, int out_size, void* d_ws, size_t ws_size,
                              hipStream_t stream) {
    (void)in_sizes; (void)n_in; (void)out_size; (void)d_ws; (void)ws_size;
    const float* x      = (const float*)d_in[0];
    const float* coeffs = (const float*)d_in[1];
    float* out          = (float*)d_out;

    dim3 grid(NBLOCKS), block(TPB);
    hipLaunchKernelGGL(fourier_kan_kernel, grid, block, 0, stream, x, coeffs, out);
}